// LossFunction_ncc_68539088110075
// MI455X (gfx1250) — compile-verified
//
#include <hip/hip_runtime.h>

typedef __attribute__((ext_vector_type(2))) float v2f;
typedef __attribute__((ext_vector_type(8))) float v8f;

#define NB_RED 512
#define TPB 256

__device__ __forceinline__ int imin(int a, int b) { return a < b ? a : b; }
__device__ __forceinline__ int imax(int a, int b) { return a > b ? a : b; }

__device__ __forceinline__ float block_reduce_sum(float v, float* sm) {
  int t = threadIdx.x;
  sm[t] = v;
  __syncthreads();
  for (int s = TPB >> 1; s > 0; s >>= 1) {
    if (t < s) sm[t] += sm[t + s];
    __syncthreads();
  }
  float r = sm[0];
  __syncthreads();
  return r;
}

// ---------------- gradient_loss partial sums --------------------------------
// slots: 3 = dx (H axis), 4 = dy (D axis), 5 = dz (W axis)
__global__ void grad_kernel(const float* __restrict__ flow, float* __restrict__ partials,
                            int D, int H, int W) {
  __shared__ float sm[TPB];
  const long long N = (long long)D * H * W;
  const long long total = 3LL * N;
  const long long stride = (long long)gridDim.x * blockDim.x;
  float sdx = 0.f, sdy = 0.f, sdz = 0.f;
  for (long long e = (long long)blockIdx.x * blockDim.x + threadIdx.x; e < total; e += stride) {
    long long r = e % N;
    int d = (int)(r / ((long long)H * W));
    int h = (int)((r / W) % H);
    int w = (int)(r % W);
    float x0 = flow[e];
    if (d < D - 1) { float df = flow[e + (long long)H * W] - x0; sdy += df * df; }
    if (h < H - 1) { float df = flow[e + W] - x0;                sdx += df * df; }
    if (w < W - 1) { float df = flow[e + 1] - x0;                sdz += df * df; }
  }
  float bx = block_reduce_sum(sdx, sm);
  float by = block_reduce_sum(sdy, sm);
  float bz = block_reduce_sum(sdz, sm);
  if (threadIdx.x == 0) {
    partials[3 * NB_RED + blockIdx.x] = bx;
    partials[4 * NB_RED + blockIdx.x] = by;
    partials[5 * NB_RED + blockIdx.x] = bz;
  }
}

// ---------------- trilinear resize (align_corners=True, clamp) * factor -----
__global__ void resize_kernel(float* __restrict__ dst, const float* __restrict__ src,
                              int Di, int Hi, int Wi, int Do, int Ho, int Wo, float factor) {
  int idx = blockIdx.x * blockDim.x + threadIdx.x;
  int N = Do * Ho * Wo;
  if (idx >= N) return;
  int w = idx % Wo;
  int h = (idx / Wo) % Ho;
  int d = idx / (Wo * Ho);
  float fd = (float)d * (float)(Di - 1) / (float)(Do - 1);
  float fh = (float)h * (float)(Hi - 1) / (float)(Ho - 1);
  float fw = (float)w * (float)(Wi - 1) / (float)(Wo - 1);
  int d0 = imin((int)fd, Di - 1), h0 = imin((int)fh, Hi - 1), w0 = imin((int)fw, Wi - 1);
  int d1 = imin(d0 + 1, Di - 1), h1 = imin(h0 + 1, Hi - 1), w1 = imin(w0 + 1, Wi - 1);
  float td = fd - (float)d0, th = fh - (float)h0, tw = fw - (float)w0;
  const long long sD = (long long)Hi * Wi;
  #define SRC(di, hi, wi) src[(long long)(di) * sD + (long long)(hi) * Wi + (wi)]
  float c00 = SRC(d0, h0, w0) * (1.f - tw) + SRC(d0, h0, w1) * tw;
  float c01 = SRC(d0, h1, w0) * (1.f - tw) + SRC(d0, h1, w1) * tw;
  float c10 = SRC(d1, h0, w0) * (1.f - tw) + SRC(d1, h0, w1) * tw;
  float c11 = SRC(d1, h1, w0) * (1.f - tw) + SRC(d1, h1, w1) * tw;
  #undef SRC
  float c0 = c00 * (1.f - th) + c01 * th;
  float c1 = c10 * (1.f - th) + c11 * th;
  dst[idx] = factor * (c0 * (1.f - td) + c1 * td);
}

// ---------------- warp: trilinear sample at grid+flow, zero padding ---------
__global__ void warp_kernel(float* __restrict__ dst, const float* __restrict__ src,
                            const float* __restrict__ flow, int D, int H, int W) {
  int idx = blockIdx.x * blockDim.x + threadIdx.x;
  int N = D * H * W;
  if (idx >= N) return;
  int w = idx % W;
  int h = (idx / W) % H;
  int d = idx / (W * H);
  float fd = (float)d + flow[idx];
  float fh = (float)h + flow[N + idx];
  float fw = (float)w + flow[2 * N + idx];
  float dd = floorf(fd), hh = floorf(fh), ww = floorf(fw);
  int d0 = (int)dd, h0 = (int)hh, w0 = (int)ww;
  float td = fd - dd, th = fh - hh, tw = fw - ww;
  float acc = 0.f;
  #pragma unroll
  for (int cd = 0; cd < 2; ++cd)
    #pragma unroll
    for (int ch = 0; ch < 2; ++ch)
      #pragma unroll
      for (int cw = 0; cw < 2; ++cw) {
        int di = d0 + cd, hi = h0 + ch, wi = w0 + cw;
        if (di >= 0 && di < D && hi >= 0 && hi < H && wi >= 0 && wi < W) {
          float wt = (cd ? td : 1.f - td) * (ch ? th : 1.f - th) * (cw ? tw : 1.f - tw);
          acc += wt * src[((long long)di * H + hi) * W + wi];
        }
      }
  dst[idx] = acc;
}

// ---------------- box pass along W: build 5 channels ------------------------
__global__ void box_x_kernel(float* __restrict__ buf, const float* __restrict__ I,
                             const float* __restrict__ J, int D, int H, int W, int p) {
  int idx = blockIdx.x * blockDim.x + threadIdx.x;
  int N = D * H * W;
  if (idx >= N) return;
  int w = idx % W;
  int base = idx - w;
  int lo = imax(w - p, 0), hi = imin(w + p, W - 1);
  float sI = 0.f, sJ = 0.f, sII = 0.f, sJJ = 0.f, sIJ = 0.f;
  for (int t = lo; t <= hi; ++t) {
    float iv = I[base + t], jv = J[base + t];
    sI += iv; sJ += jv; sII += iv * iv; sJJ += jv * jv; sIJ += iv * jv;
  }
  buf[idx] = sI;
  buf[N + idx] = sJ;
  buf[2 * N + idx] = sII;
  buf[3 * N + idx] = sJJ;
  buf[4 * N + idx] = sIJ;
}

// ---------------- box pass along H: in-place via LDS tile (H x 32) ----------
__global__ void box_y_kernel(float* __restrict__ buf, int D, int H, int W, int p) {
  extern __shared__ float lds[]; // H * 32 floats
  int nwc = (W + 31) / 32;
  int d = blockIdx.x / nwc;
  int wc = (blockIdx.x % nwc) * 32;
  int N = D * H * W;
  int tileN = H * 32;
  for (int c = 0; c < 5; ++c) {
    float* plane = buf + c * N + d * H * W;
    for (int i = threadIdx.x; i < tileN; i += blockDim.x) {
      int h = i >> 5, wl = i & 31, w = wc + wl;
      lds[i] = (w < W) ? plane[h * W + w] : 0.f;
    }
    __syncthreads();
    for (int i = threadIdx.x; i < tileN; i += blockDim.x) {
      int h = i >> 5, wl = i & 31, w = wc + wl;
      if (w < W) {
        int lo = imax(h - p, 0), hi = imin(h + p, H - 1);
        float s = 0.f;
        for (int t = lo; t <= hi; ++t) s += lds[(t << 5) + wl];
        plane[h * W + w] = s;
      }
    }
    __syncthreads();
  }
}

// ---------------- box pass along D + cc + block reduction -------------------
__global__ void box_z_cc_kernel(const float* __restrict__ buf, float* __restrict__ partials,
                                int slot, int D, int H, int W, int p, float win) {
  __shared__ float sm[TPB];
  int N = D * H * W;
  float invWin = 1.0f / win;
  float acc = 0.f;
  int stride = gridDim.x * blockDim.x;
  for (int idx = blockIdx.x * blockDim.x + threadIdx.x; idx < N; idx += stride) {
    int hw = idx % (H * W);
    int d = idx / (H * W);
    int lo = imax(d - p, 0), hi = imin(d + p, D - 1);
    float sI = 0.f, sJ = 0.f, sII = 0.f, sJJ = 0.f, sIJ = 0.f;
    for (int t = lo; t <= hi; ++t) {
      int off = t * H * W + hw;
      sI += buf[off];
      sJ += buf[N + off];
      sII += buf[2 * N + off];
      sJJ += buf[3 * N + off];
      sIJ += buf[4 * N + off];
    }
    float cross = sIJ - sI * sJ * invWin;
    float Ivar = sII - sI * sI * invWin;
    float Jvar = sJJ - sJ * sJ * invWin;
    acc += cross * cross / (Ivar * Jvar + 1e-5f);
  }
  float b = block_reduce_sum(acc, sm);
  if (threadIdx.x == 0) partials[slot * NB_RED + blockIdx.x] = b;
}

// ---------------- final: WMMA ones-matmul reduction of 512 partials ---------
// A = ones(16x4)  =>  D[i][j] = sum_k B[k][j] + C[i][j]; the total over all
// 256 entries of D is 16x the sum of every float packed into B, independent
// of the exact operand VGPR layout.
__device__ float wave_sum512(const float* __restrict__ p) {
  int lane = threadIdx.x; // single wave of 32, EXEC all ones
#if __has_builtin(__builtin_amdgcn_wmma_f32_16x16x4_f32)
  v2f a; a.x = 1.0f; a.y = 1.0f;
  v8f c = {};
  for (int i = 0; i < NB_RED; i += 64) {
    v2f b; b.x = p[i + lane]; b.y = p[i + 32 + lane];
    c = __builtin_amdgcn_wmma_f32_16x16x4_f32(false, a, false, b, (short)0, c, false, false);
  }
  float s = c[0] + c[1] + c[2] + c[3] + c[4] + c[5] + c[6] + c[7];
  for (int off = 16; off >= 1; off >>= 1) s += __shfl_xor(s, off, 32);
  return s * (1.0f / 16.0f);
#else
  float s = 0.f;
  for (int i = lane; i < NB_RED; i += 32) s += p[i];
  for (int off = 16; off >= 1; off >>= 1) s += __shfl_xor(s, off, 32);
  return s;
#endif
}

__global__ void final_kernel(float* __restrict__ out, const float* __restrict__ partials,
                             const float* __restrict__ h1, const float* __restrict__ h2,
                             const float* __restrict__ h3, const float* __restrict__ h4) {
  float scc0 = wave_sum512(partials + 0 * NB_RED);
  float scc1 = wave_sum512(partials + 1 * NB_RED);
  float scc2 = wave_sum512(partials + 2 * NB_RED);
  float sdx  = wave_sum512(partials + 3 * NB_RED);
  float sdy  = wave_sum512(partials + 4 * NB_RED);
  float sdz  = wave_sum512(partials + 5 * NB_RED);
  if (threadIdx.x == 0) {
    const float N0 = 160.f * 192.f * 224.f;
    const float N1 = 80.f * 96.f * 112.f;
    const float N2 = 40.f * 48.f * 56.f;
    float ncc0 = -scc0 / N0;
    float ncc1 = -scc1 / N1;
    float ncc2 = -scc2 / N2;
    const float cdx = 3.f * 160.f * 191.f * 224.f; // H-axis diffs
    const float cdy = 3.f * 159.f * 192.f * 224.f; // D-axis diffs
    const float cdz = 3.f * 160.f * 192.f * 223.f; // W-axis diffs
    float grad = (sdx / cdx + sdy / cdy + sdz / cdz) * (1.0f / 3.0f);
    float multi = (*h3) * ncc1 + (*h4) * ncc2;
    float loss = multi + (*h1) * ncc0 + (*h2) * grad;
    out[0] = loss;
    out[1] = ncc0;
    out[2] = grad;
  }
}

extern "C" void kernel_launch(void* const* d_in, const int* in_sizes, int n_in,
                              void* d_out, int out_size, void* d_ws, size_t ws_size,
                              hipStream_t stream) {
  const float* y    = (const float*)d_in[0];
  const float* tgt  = (const float*)d_in[1];
  const float* src  = (const float*)d_in[2];
  const float* flow = (const float*)d_in[3];
  const float* if1  = (const float*)d_in[4];
  const float* if2  = (const float*)d_in[6];
  const float* h1   = (const float*)d_in[8];
  const float* h2   = (const float*)d_in[9];
  const float* h3   = (const float*)d_in[10];
  const float* h4   = (const float*)d_in[11];
  float* out = (float*)d_out;

  const int D0 = 160, H0 = 192, W0 = 224, N0 = D0 * H0 * W0;
  const int D1 = 80,  H1 = 96,  W1 = 112, N1 = D1 * H1 * W1;
  const int D2 = 40,  H2 = 48,  W2 = 56,  N2 = D2 * H2 * W2;

  float* partials = (float*)d_ws;                     // 6 * 512 floats
  float* R = (float*)((char*)d_ws + 16384);           // overlay region

  // gradient_loss partials (slots 3..5)
  grad_kernel<<<NB_RED, TPB, 0, stream>>>(flow, partials, D0, H0, W0);

  // -------- full-res NCC(tgt, y, 9) --------
  float* bufF = R; // 5 * N0 floats
  box_x_kernel<<<(N0 + TPB - 1) / TPB, TPB, 0, stream>>>(bufF, tgt, y, D0, H0, W0, 4);
  box_y_kernel<<<D0 * ((W0 + 31) / 32), TPB, (size_t)H0 * 32 * sizeof(float), stream>>>(bufF, D0, H0, W0, 4);
  box_z_cc_kernel<<<NB_RED, TPB, 0, stream>>>(bufF, partials, 0, D0, H0, W0, 4, 729.0f);

  // -------- half-res: resize, warp, NCC(warped1, z1_tgt, 7) --------
  float* z1s  = R;
  float* z1t  = R + N1;
  float* w1   = R + 2 * N1;
  float* bufH = R + 3 * N1; // 5 * N1
  resize_kernel<<<(N1 + TPB - 1) / TPB, TPB, 0, stream>>>(z1t, tgt, D0, H0, W0, D1, H1, W1, 0.5f);
  resize_kernel<<<(N1 + TPB - 1) / TPB, TPB, 0, stream>>>(z1s, src, D0, H0, W0, D1, H1, W1, 0.5f);
  warp_kernel<<<(N1 + TPB - 1) / TPB, TPB, 0, stream>>>(w1, z1s, if1, D1, H1, W1);
  box_x_kernel<<<(N1 + TPB - 1) / TPB, TPB, 0, stream>>>(bufH, w1, z1t, D1, H1, W1, 3);
  box_y_kernel<<<D1 * ((W1 + 31) / 32), TPB, (size_t)H1 * 32 * sizeof(float), stream>>>(bufH, D1, H1, W1, 3);
  box_z_cc_kernel<<<NB_RED, TPB, 0, stream>>>(bufH, partials, 1, D1, H1, W1, 3, 343.0f);

  // -------- quarter-res: resize, warp, NCC(warped2, z2_tgt, 5) --------
  float* Q = R + 8 * N1;
  float* z2s  = Q;
  float* z2t  = Q + N2;
  float* w2   = Q + 2 * N2;
  float* bufQ = Q + 3 * N2; // 5 * N2
  resize_kernel<<<(N2 + TPB - 1) / TPB, TPB, 0, stream>>>(z2t, tgt, D0, H0, W0, D2, H2, W2, 0.25f);
  resize_kernel<<<(N2 + TPB - 1) / TPB, TPB, 0, stream>>>(z2s, src, D0, H0, W0, D2, H2, W2, 0.25f);
  warp_kernel<<<(N2 + TPB - 1) / TPB, TPB, 0, stream>>>(w2, z2s, if2, D2, H2, W2);
  box_x_kernel<<<(N2 + TPB - 1) / TPB, TPB, 0, stream>>>(bufQ, w2, z2t, D2, H2, W2, 2);
  box_y_kernel<<<D2 * ((W2 + 31) / 32), TPB, (size_t)H2 * 32 * sizeof(float), stream>>>(bufQ, D2, H2, W2, 2);
  box_z_cc_kernel<<<NB_RED, TPB, 0, stream>>>(bufQ, partials, 2, D2, H2, W2, 2, 125.0f);

  // -------- final combine (single wave, WMMA reduction) --------
  final_kernel<<<1, 32, 0, stream>>>(out, partials, h1, h2, h3, h4);
}